// MessageArMLP_2156073583069
// MI455X (gfx1250) — compile-verified
//
#include <hip/hip_runtime.h>
#include <hip/hip_bf16.h>

// ---------------------------------------------------------------------------
// Problem constants (from the reference)
// ---------------------------------------------------------------------------
#define RADIAL_DIM   8     // R
#define ANGULAR_DIM  20    // A
#define CHANNEL      8     // C
#define N_GROUPS     4     // G
#define RADIAL_EMB   8     // I (contraction dim)
#define FEAT_PER_NODE (RADIAL_DIM * ANGULAR_DIM * CHANNEL)   // 1280 floats
#define DECAY_STRIDE  (N_GROUPS * CHANNEL)                   // 32 floats per edge

typedef __attribute__((ext_vector_type(2))) float v2f;
typedef __attribute__((ext_vector_type(4))) float v4f;   // native vector for NT builtins
typedef __attribute__((ext_vector_type(8))) float v8f;

__device__ __forceinline__ float fast_sigmoid(float x) {
  // v_exp_f32 + v_add_f32 + v_rcp_f32 (avoid the IEEE-div expansion)
  return __builtin_amdgcn_rcpf(1.0f + __expf(-x));
}

// ---------------------------------------------------------------------------
// Kernel 1: decay[e][g][c] = sigmoid(rc[e,:] . W[g,:,c]) * cutoff[e]
//
// WMMA mapping (V_WMMA_F32_16X16X4_F32, wave32, one wave per block):
//   - M (=16)   : 16 edges per tile
//   - N (=16)   : two groups packed side by side (cols 0-7 = group 2p+0
//                 channels, cols 8-15 = group 2p+1 channels)
//   - K (=4 x2) : RADIAL_EMB=8 -> two chained WMMAs (K 0-3, K 4-7)
//
// 32-bit A 16x4 layout (ISA 7.12.2): lane L (half h=L/16, m=L%16) holds
//   {A[m][2h], A[m][2h+1]}; B 4x16 analogous with N = lane%16.
// C/D 16x16 f32 layout: vgpr v of lane L holds D[v + 8h][L%16].
// ---------------------------------------------------------------------------
__global__ void __launch_bounds__(32)
decay_wmma_kernel(const float* __restrict__ rc,      // [E, 8]
                  const float* __restrict__ cutoff,  // [E]
                  const float* __restrict__ W,       // [4, 8, 8]  (g, i, c)
                  float* __restrict__ decay,         // [E, 4, 8]
                  int E) {
  const int tile = blockIdx.x * 16;
  const int lane = threadIdx.x;        // 0..31
  const int h    = lane >> 4;          // lane half
  const int m    = lane & 15;          // edge-in-tile for A; column N for B/D
  const bool full = (tile + 16 <= E);  // wave-uniform tail check

  // --- A fragments: rc rows for this tile (clamped tail reads) ---
  int ea = tile + m; if (ea >= E) ea = E - 1;
  const float* rcrow = rc + (size_t)ea * RADIAL_EMB;
  v2f a0, a1;
  a0.x = rcrow[2 * h + 0];  a0.y = rcrow[2 * h + 1];   // K = 0..3
  a1.x = rcrow[2 * h + 4];  a1.y = rcrow[2 * h + 5];   // K = 4..7

  // --- cutoff for the 8 edges this lane's D rows correspond to ---
  float cf[8];
#pragma unroll
  for (int v = 0; v < 8; ++v) {
    int ec = tile + v + 8 * h; if (ec >= E) ec = E - 1;
    cf[v] = cutoff[ec];
  }

  const int gsub = m >> 3;     // group within the pair (from column N)
  const int ch   = m & 7;      // channel (from column N)

#pragma unroll
  for (int pair = 0; pair < 2; ++pair) {
    const int g = pair * 2 + gsub;
    // --- B fragment: W[g][k][ch], k striped like A's K ---
    const float* wg = W + g * (RADIAL_EMB * CHANNEL) + ch;   // stride 8 over k
    v2f b0, b1;
    b0.x = wg[(2 * h + 0) * CHANNEL];  b0.y = wg[(2 * h + 1) * CHANNEL];
    b1.x = wg[(2 * h + 4) * CHANNEL];  b1.y = wg[(2 * h + 5) * CHANNEL];

    v8f acc = {};
    // 8 args: (neg_a, A, neg_b, B, c_mod, C, reuse_a, reuse_b)
    acc = __builtin_amdgcn_wmma_f32_16x16x4_f32(false, a0, false, b0,
                                                (short)0, acc, false, false);
    acc = __builtin_amdgcn_wmma_f32_16x16x4_f32(false, a1, false, b1,
                                                (short)0, acc, false, false);

    // --- epilogue: sigmoid * cutoff, scatter to decay[e][g][ch] ---
    float* dbase = decay + (size_t)(tile + 8 * h) * DECAY_STRIDE
                         + g * CHANNEL + ch;
    if (full) {
      // branch-free common path (E is a multiple of 16 in practice)
#pragma unroll
      for (int v = 0; v < 8; ++v) {
        dbase[(size_t)v * DECAY_STRIDE] = fast_sigmoid(acc[v]) * cf[v];
      }
    } else {
#pragma unroll
      for (int v = 0; v < 8; ++v) {
        if (tile + v + 8 * h < E)
          dbase[(size_t)v * DECAY_STRIDE] = fast_sigmoid(acc[v]) * cf[v];
      }
    }
  }
}

// ---------------------------------------------------------------------------
// Kernel 2: message[e][r][a][c] = node_feat[src[e]][r][a][c] * decay[e][g(a)][c]
// One 320-thread block per edge; one v4f per thread (1280 floats / edge).
// Streaming: global_load_b128 -> v_mul -> nontemporal global_store_b128.
// node_feat (51 MB) is L2-resident; NT stores keep the 614 MB output from
// evicting it.
// ---------------------------------------------------------------------------
__global__ void __launch_bounds__(320)
message_kernel(const float* __restrict__ node_feat, // [N, 8, 20, 8]
               const int*   __restrict__ edge_src,  // edge_index row 0, [E]
               const float* __restrict__ decay,     // [E, 4, 8]
               float*       __restrict__ out) {     // [E, 8, 20, 8]
  const int e   = blockIdx.x;
  const int t   = threadIdx.x;       // 0..319
  const int r   = t / 40;            // 0..7
  const int rem = t - r * 40;
  const int a   = rem >> 1;          // 0..19
  const int c4  = (rem & 1) * 4;     // 0 or 4

  // angular index -> l-group: [0,1)=0, [1,4)=1, [4,10)=2, [10,20)=3
  const int g = (a >= 10) ? 3 : (a >= 4) ? 2 : (a >= 1) ? 1 : 0;

  const int src = edge_src[e];       // wave-uniform -> scalar load
  const size_t ra8 = (size_t)(r * ANGULAR_DIM + a) * CHANNEL + c4;

  const v4f nf = *(const v4f*)(node_feat + (size_t)src * FEAT_PER_NODE + ra8);
  const v4f dc = *(const v4f*)(decay + (size_t)e * DECAY_STRIDE
                                     + g * CHANNEL + c4);
  const v4f o = nf * dc;
  __builtin_nontemporal_store(o, (v4f*)(out + (size_t)e * FEAT_PER_NODE + ra8));
}

// ---------------------------------------------------------------------------
// Launch. Input order (setup_inputs dict):
//   d_in[0] node_feat          f32 [N, 8, 20, 8]
//   d_in[1] radial_component   f32 [E, 8]
//   d_in[2] radial_cutoff_fn   f32 [E]
//   d_in[3] weights            f32 [4, 8, 8]
//   d_in[4] edge_index         int [2, E]   (row 0 = senders)
// ---------------------------------------------------------------------------
extern "C" void kernel_launch(void* const* d_in, const int* in_sizes, int n_in,
                              void* d_out, int out_size, void* d_ws, size_t ws_size,
                              hipStream_t stream) {
  const float* node_feat = (const float*)d_in[0];
  const float* rc        = (const float*)d_in[1];
  const float* cutoff    = (const float*)d_in[2];
  const float* weights   = (const float*)d_in[3];
  const int*   edge_src  = (const int*)d_in[4];   // first E entries = row 0

  const int E = in_sizes[2];                      // radial_cutoff_fn has E elems
  float* decay = (float*)d_ws;                    // [E, 4, 8] = E*128 bytes

  const int tiles = (E + 15) / 16;
  decay_wmma_kernel<<<tiles, 32, 0, stream>>>(rc, cutoff, weights, decay, E);
  message_kernel<<<E, 320, 0, stream>>>(node_feat, edge_src, decay,
                                        (float*)d_out);
}